// ROIAwareGCN_48077863911624
// MI455X (gfx1250) — compile-verified
//
#include <hip/hip_runtime.h>
#include <hip/hip_bf16.h>

#define N_NODES 50000
#define N_EDGES 600000

typedef __attribute__((ext_vector_type(2))) float v2f;
typedef __attribute__((ext_vector_type(8))) float v8f;

// ---------------------------------------------------------------------------
// Degree / normalization
// ---------------------------------------------------------------------------
__global__ void init_deg_kernel(float* deg, int n) {
    int i = blockIdx.x * blockDim.x + threadIdx.x;
    if (i < n) deg[i] = 1.0f;  // self-loop contributes 1 to its own dst
}

__global__ void accum_deg_kernel(const int* __restrict__ ei, float* deg, int nE) {
    int i = blockIdx.x * blockDim.x + threadIdx.x;
    if (i < nE) atomicAdd(&deg[ei[nE + i]], 1.0f);  // dst row of edge_index
}

__global__ void dinv_kernel(const float* __restrict__ deg, float* dinv, int n) {
    int i = blockIdx.x * blockDim.x + threadIdx.x;
    if (i < n) {
        float d = deg[i];
        dinv[i] = (d > 0.0f) ? __frsqrt_rn(d) : 0.0f;
    }
}

// ---------------------------------------------------------------------------
// Dense GEMM  C[M x N] = A[M x 128] @ W[128 x N]  using V_WMMA_F32_16X16X4_F32
// One 16x16 C-tile per wave; NCOLS/16 waves per block; K loop = 32 wmma steps.
// ---------------------------------------------------------------------------
template <int NCOLS>
__global__ __launch_bounds__(256) void gemm_wmma_kernel(
    const float* __restrict__ A, const float* __restrict__ W,
    float* __restrict__ C, int M) {
    constexpr int K = 128;
    __shared__ float As[16][K + 4];  // +4 pad breaks LDS bank conflicts

    const int mbase = blockIdx.x * 16;
    const int tid = threadIdx.x;

    // Cooperative load of the 16 x 128 A tile (zero-pad past M).
    for (int i = tid; i < 16 * K; i += blockDim.x) {
        int r = i >> 7, c = i & 127;
        int m = mbase + r;
        As[r][c] = (m < M) ? A[(long long)m * K + c] : 0.0f;
    }
    __syncthreads();

    const int wave = tid >> 5;
    const int lane = tid & 31;
    const int ncol = wave * 16 + (lane & 15);
    const int khalf = (lane >> 4) << 1;  // lanes 0-15 -> k,k+1 ; lanes 16-31 -> k+2,k+3
    const int arow = lane & 15;

    // Preload all B fragments (weights are L2-resident; offsets are immediates).
    v2f bfrag[K / 4];
#pragma unroll
    for (int kk = 0; kk < K / 4; ++kk) {
        int ko = kk * 4 + khalf;
        bfrag[kk].x = W[ko * NCOLS + ncol];
        bfrag[kk].y = W[(ko + 1) * NCOLS + ncol];
    }

    v8f acc = {};
#pragma unroll
    for (int kk = 0; kk < K / 4; ++kk) {
        int ko = kk * 4 + khalf;
        v2f a;
        a.x = As[arow][ko];
        a.y = As[arow][ko + 1];
        acc = __builtin_amdgcn_wmma_f32_16x16x4_f32(
            /*neg_a=*/false, a, /*neg_b=*/false, bfrag[kk],
            /*c_mod=*/(short)0, acc, /*reuse_a=*/false, /*reuse_b=*/false);
    }

    // C/D layout: VGPR r -> row r (lanes 0-15) / r+8 (lanes 16-31), col = lane%16.
    const int mrow = mbase + ((lane >> 4) << 3);
#pragma unroll
    for (int r = 0; r < 8; ++r) {
        int m = mrow + r;
        if (m < M) C[(long long)m * NCOLS + ncol] = acc[r];
    }
}

// ---------------------------------------------------------------------------
// Aggregation: out = bias (init), then out[dst] += norm * h[src] over edges
// and self-loops.  One wave covers one edge's channels (float4 per thread).
// ---------------------------------------------------------------------------
__global__ void init_bias_kernel(float* __restrict__ out, const float* __restrict__ bias,
                                 int n, int lgC) {
    int i = blockIdx.x * blockDim.x + threadIdx.x;
    if (i < n) out[i] = bias[i & ((1 << lgC) - 1)];
}

__global__ void scatter_kernel(const float* __restrict__ h, const int* __restrict__ ei,
                               const float* __restrict__ dinv, float* __restrict__ out,
                               int nE, int nN, int lgC) {
    const int Cch = 1 << lgC;
    const int lgG = lgC - 2;              // float4 groups per row
    const long long total = (long long)(nE + nN) << lgG;
    const long long stride = (long long)gridDim.x * blockDim.x;
    for (long long i = (long long)blockIdx.x * blockDim.x + threadIdx.x; i < total;
         i += stride) {
        long long e = i >> lgG;
        int cg = (int)(i & ((1 << lgG) - 1));
        int s, d;
        float nm;
        if (e < nE) {
            s = ei[e];
            d = ei[nE + e];
            nm = dinv[s] * dinv[d];
        } else {
            int v = (int)(e - nE);
            s = v;
            d = v;
            float di = dinv[v];
            nm = di * di;
        }
        const float4 v4 = ((const float4*)(h + (long long)s * Cch))[cg];
        float* o = out + (long long)d * Cch + (cg << 2);
        atomicAdd(o + 0, nm * v4.x);
        atomicAdd(o + 1, nm * v4.y);
        atomicAdd(o + 2, nm * v4.z);
        atomicAdd(o + 3, nm * v4.w);
    }
}

__global__ void relu_kernel(float* __restrict__ x, int n) {
    int i = blockIdx.x * blockDim.x + threadIdx.x;
    if (i < n) x[i] = fmaxf(x[i], 0.0f);
}

__global__ void zero_kernel(float* __restrict__ x, int n) {
    int i = blockIdx.x * blockDim.x + threadIdx.x;
    if (i < n) x[i] = 0.0f;
}

// ---------------------------------------------------------------------------
// Global mean pool (sums + counts via atomics)
// ---------------------------------------------------------------------------
__global__ void pool_kernel(const float* __restrict__ G, const int* __restrict__ batch,
                            float* __restrict__ sums, float* __restrict__ cnts, int nN) {
    int i = blockIdx.x * blockDim.x + threadIdx.x;
    int node = i >> 6;
    int c = i & 63;
    if (node >= nN) return;
    int b = batch[node];
    atomicAdd(&sums[b * 64 + c], G[(long long)node * 64 + c]);
    if (c == 0) atomicAdd(&cnts[b], 1.0f);
}

// ---------------------------------------------------------------------------
// Head MLP: one block (64 threads) per graph.  z = [pooled(64), demo(8)]
// ---------------------------------------------------------------------------
__global__ void head_kernel(const float* __restrict__ sums, const float* __restrict__ cnts,
                            const float* __restrict__ demo, const float* __restrict__ Wf1,
                            const float* __restrict__ bf1, const float* __restrict__ Wf2,
                            const float* __restrict__ bf2, const float* __restrict__ Wf3,
                            const float* __restrict__ bf3, float* __restrict__ out) {
    __shared__ float z[72];
    __shared__ float z1[64];
    __shared__ float z2[32];
    const int g = blockIdx.x;
    const int t = threadIdx.x;
    const float c = fmaxf(cnts[g], 1.0f);
    z[t] = sums[g * 64 + t] / c;
    if (t < 8) z[64 + t] = demo[g * 8 + t];
    __syncthreads();
    float a1 = bf1[t];
    for (int i = 0; i < 72; ++i) a1 += z[i] * Wf1[i * 64 + t];
    z1[t] = fmaxf(a1, 0.0f);
    __syncthreads();
    if (t < 32) {
        float a2 = bf2[t];
        for (int i = 0; i < 64; ++i) a2 += z1[i] * Wf2[i * 32 + t];
        z2[t] = fmaxf(a2, 0.0f);
    }
    __syncthreads();
    if (t < 2) {
        float a3 = bf3[t];
        for (int i = 0; i < 32; ++i) a3 += z2[i] * Wf3[i * 2 + t];
        out[g * 2 + t] = a3;
    }
}

// ---------------------------------------------------------------------------
extern "C" void kernel_launch(void* const* d_in, const int* in_sizes, int n_in,
                              void* d_out, int out_size, void* d_ws, size_t ws_size,
                              hipStream_t stream) {
    (void)in_sizes; (void)n_in; (void)out_size; (void)ws_size;
    const float* x    = (const float*)d_in[0];
    const int*   ei   = (const int*)d_in[1];
    const int*   batch= (const int*)d_in[2];
    const float* demo = (const float*)d_in[3];
    const float* W1 = (const float*)d_in[4];
    const float* b1 = (const float*)d_in[5];
    const float* W2 = (const float*)d_in[6];
    const float* b2 = (const float*)d_in[7];
    const float* W3 = (const float*)d_in[8];
    const float* b3 = (const float*)d_in[9];
    const float* Wf1 = (const float*)d_in[10];
    const float* bf1 = (const float*)d_in[11];
    const float* Wf2 = (const float*)d_in[12];
    const float* bf2 = (const float*)d_in[13];
    const float* Wf3 = (const float*)d_in[14];
    const float* bf3 = (const float*)d_in[15];
    float* out = (float*)d_out;

    const int NN = N_NODES;
    const int NE = N_EDGES;
    const int NG = 100;

    // Workspace carving (256B aligned)
    char* ws = (char*)d_ws;
    size_t off = 0;
    auto carve = [&](size_t bytes) {
        void* p = ws + off;
        off = (off + bytes + 255) & ~(size_t)255;
        return p;
    };
    float* T    = (float*)carve((size_t)NN * 128 * sizeof(float));  // GEMM output
    float* G    = (float*)carve((size_t)NN * 128 * sizeof(float));  // aggregation
    float* deg  = (float*)carve((size_t)NN * sizeof(float));
    float* dinv = (float*)carve((size_t)NN * sizeof(float));
    float* sums = (float*)carve((size_t)NG * 64 * sizeof(float));
    float* cnts = (float*)carve((size_t)NG * sizeof(float));

    const int TB = 256;
    const int gemmBlocks = (NN + 15) / 16;  // 3125
    const int scatterBlocks = 4096;

    // ---- degree & symmetric normalization
    init_deg_kernel<<<(NN + TB - 1) / TB, TB, 0, stream>>>(deg, NN);
    accum_deg_kernel<<<(NE + TB - 1) / TB, TB, 0, stream>>>(ei, deg, NE);
    dinv_kernel<<<(NN + TB - 1) / TB, TB, 0, stream>>>(deg, dinv, NN);

    // ---- layer 1: 128 -> 128
    gemm_wmma_kernel<128><<<gemmBlocks, 256, 0, stream>>>(x, W1, T, NN);
    init_bias_kernel<<<(NN * 128 + TB - 1) / TB, TB, 0, stream>>>(G, b1, NN * 128, 7);
    scatter_kernel<<<scatterBlocks, TB, 0, stream>>>(T, ei, dinv, G, NE, NN, 7);
    relu_kernel<<<(NN * 128 + TB - 1) / TB, TB, 0, stream>>>(G, NN * 128);

    // ---- layer 2: 128 -> 128
    gemm_wmma_kernel<128><<<gemmBlocks, 256, 0, stream>>>(G, W2, T, NN);
    init_bias_kernel<<<(NN * 128 + TB - 1) / TB, TB, 0, stream>>>(G, b2, NN * 128, 7);
    scatter_kernel<<<scatterBlocks, TB, 0, stream>>>(T, ei, dinv, G, NE, NN, 7);
    relu_kernel<<<(NN * 128 + TB - 1) / TB, TB, 0, stream>>>(G, NN * 128);

    // ---- layer 3: 128 -> 64
    gemm_wmma_kernel<64><<<gemmBlocks, 128, 0, stream>>>(G, W3, T, NN);
    init_bias_kernel<<<(NN * 64 + TB - 1) / TB, TB, 0, stream>>>(G, b3, NN * 64, 6);
    scatter_kernel<<<scatterBlocks, TB, 0, stream>>>(T, ei, dinv, G, NE, NN, 6);
    relu_kernel<<<(NN * 64 + TB - 1) / TB, TB, 0, stream>>>(G, NN * 64);

    // ---- mean pool + head
    zero_kernel<<<(NG * 64 + TB - 1) / TB, TB, 0, stream>>>(sums, NG * 64);
    zero_kernel<<<1, TB, 0, stream>>>(cnts, NG);
    pool_kernel<<<(NN * 64 + TB - 1) / TB, TB, 0, stream>>>(G, batch, sums, cnts, NN);
    head_kernel<<<NG, 64, 0, stream>>>(sums, cnts, demo, Wf1, bf1, Wf2, bf2, Wf3, bf3, out);
}